// GNNLayer_36661840839017
// MI455X (gfx1250) — compile-verified
//
#include <hip/hip_runtime.h>
#include <hip/hip_bf16.h>

typedef float v2f __attribute__((ext_vector_type(2)));
typedef float v8f __attribute__((ext_vector_type(8)));

#define IN_DIM   64
#define OUT_DIM  64
#define ATTN_DIM 32

// ---------------------------------------------------------------------------
// Zero the aggregation buffer (d_out is poisoned; atomics need zeros).
// ---------------------------------------------------------------------------
__global__ void zero_f4_kernel(float4* __restrict__ p, long n4) {
    long i = (long)blockIdx.x * blockDim.x + threadIdx.x;
    if (i < n4) p[i] = make_float4(0.f, 0.f, 0.f, 0.f);
}

// ---------------------------------------------------------------------------
// Precompute attn_r[rel][k]  = rela_embed[rel] . Wr[k]          (401 x 32)
//            attn_qr[b][k]   = rela_embed[q_rel[b]] . Wqr[k] + bqr[k]  (8 x 32)
// One wave per output row; lane = attention channel k.
// ---------------------------------------------------------------------------
__global__ void attn_pre_kernel(const float* __restrict__ rela,
                                const float* __restrict__ Wr,
                                const float* __restrict__ Wqr,
                                const float* __restrict__ bqr,
                                const int*   __restrict__ q_rel,
                                float* __restrict__ attn_r,
                                float* __restrict__ attn_qr,
                                int nrel, int batch) {
    int wave = (int)((blockIdx.x * blockDim.x + threadIdx.x) >> 5);
    int k    = threadIdx.x & 31;
    if (wave >= nrel + batch) return;
    if (wave < nrel) {
        const float* hr = rela + (long)wave * IN_DIM;
        const float* w  = Wr   + (long)k    * IN_DIM;
        float s = 0.f;
        #pragma unroll
        for (int d = 0; d < IN_DIM; ++d) s += hr[d] * w[d];
        attn_r[(long)wave * ATTN_DIM + k] = s;
    } else {
        int b  = wave - nrel;
        int qr = q_rel[b];
        const float* hr = rela + (long)qr * IN_DIM;
        const float* w  = Wqr  + (long)k  * IN_DIM;
        float s = bqr[k];
        #pragma unroll
        for (int d = 0; d < IN_DIM; ++d) s += hr[d] * w[d];
        attn_qr[(long)b * ATTN_DIM + k] = s;
    }
}

// ---------------------------------------------------------------------------
// S = hidden (n_node x 64) @ Ws.T (64 x 32), exact f32 via V_WMMA_F32_16X16X4_F32.
// One wave computes one 16-row x 32-col strip; K loop = 16 steps of K=4.
// A layout: lane L holds A[M=L&15][K=2*(L>>4)+{0,1}]  -> contiguous float2.
// B layout: lane L holds B[K=2*(L>>4)+{0,1}][N=L&15] = Ws[N][K...] -> float2.
// D layout: VGPR j -> row m0 + j + 8*(L>>4), col L&15.
// ---------------------------------------------------------------------------
__global__ void node_attn_wmma_kernel(const float* __restrict__ H,
                                      const float* __restrict__ Ws,
                                      float* __restrict__ S, int n_node) {
    int wave  = (int)((blockIdx.x * blockDim.x + threadIdx.x) >> 5);
    int lane  = threadIdx.x & 31;
    int m0    = wave << 4;
    if (m0 >= n_node) return;
    int row16 = lane & 15;
    int hi    = lane >> 4;
    int arow_i = m0 + row16;
    if (arow_i >= n_node) arow_i = n_node - 1;            // safe for ragged tail
    const float* arow  = H  + (long)arow_i      * IN_DIM + 2 * hi;
    const float* b0row = Ws + (long)row16       * IN_DIM + 2 * hi;
    const float* b1row = Ws + (long)(16 + row16) * IN_DIM + 2 * hi;

    v8f c0 = {}; v8f c1 = {};
    #pragma unroll
    for (int kk = 0; kk < IN_DIM / 4; ++kk) {
        v2f a  = *(const v2f*)(arow  + 4 * kk);
        v2f b0 = *(const v2f*)(b0row + 4 * kk);
        v2f b1 = *(const v2f*)(b1row + 4 * kk);
        c0 = __builtin_amdgcn_wmma_f32_16x16x4_f32(false, a, false, b0,
                                                   (short)0, c0, false, false);
        c1 = __builtin_amdgcn_wmma_f32_16x16x4_f32(false, a, false, b1,
                                                   (short)0, c1, false, false);
    }
    #pragma unroll
    for (int j = 0; j < 8; ++j) {
        int m = m0 + j + 8 * hi;
        if (m < n_node) {
            S[(long)m * ATTN_DIM + row16]      = c0[j];
            S[(long)m * ATTN_DIM + 16 + row16] = c1[j];
        }
    }
}

// ---------------------------------------------------------------------------
// Edge pass: one wave per edge. Lane = attention channel (32) / message pair.
//   attn  = relu(S[sub] + attn_r[rel] + attn_qr[r_idx])        (32 lanes)
//   alpha = sigmoid(attn . w_alpha + b)                         (xor-reduce)
//   agg[obj] += alpha * (hidden[sub] + rela[rel])               (f32 atomics)
// All gathered rows are contiguous -> fully coalesced 128B/256B accesses.
// ---------------------------------------------------------------------------
__global__ void edge_kernel(const int*   __restrict__ edges,
                            const float* __restrict__ hidden,
                            const float* __restrict__ rela,
                            const float* __restrict__ S,
                            const float* __restrict__ attn_r,
                            const float* __restrict__ attn_qr,
                            const float* __restrict__ w_alpha_w,
                            const float* __restrict__ w_alpha_b,
                            float* __restrict__ agg, int n_edge) {
    int wave = (int)((blockIdx.x * blockDim.x + threadIdx.x) >> 5);
    int lane = threadIdx.x & 31;
    if (wave >= n_edge) return;
    const int* e = edges + (long)wave * 6;
    int r_idx = e[0];
    int rel   = e[2];
    int sub   = e[4];
    int obj   = e[5];

    float a = S[(long)sub * ATTN_DIM + lane]
            + attn_r[(long)rel * ATTN_DIM + lane]
            + attn_qr[(long)r_idx * ATTN_DIM + lane];
    a = fmaxf(a, 0.f);
    float t = a * w_alpha_w[lane];
    #pragma unroll
    for (int off = 16; off > 0; off >>= 1) t += __shfl_xor(t, off, 32);
    float alpha = 1.f / (1.f + __expf(-(t + w_alpha_b[0])));

    const float* hs = hidden + (long)sub * IN_DIM;
    const float* hr = rela   + (long)rel * IN_DIM;
    float m0v = alpha * (hs[lane]      + hr[lane]);
    float m1v = alpha * (hs[lane + 32] + hr[lane + 32]);
    float* dst = agg + (long)obj * IN_DIM;
    atomicAdd(dst + lane,      m0v);
    atomicAdd(dst + lane + 32, m1v);
}

// ---------------------------------------------------------------------------
// hidden_new = agg @ Wh.T  (n_node x 64) @ (64 x 64), IN-PLACE on agg (d_out).
// Safe: each wave owns a disjoint 16-row block; all A reads are consumed into
// accumulators before any store is issued.
// ---------------------------------------------------------------------------
__global__ void out_gemm_wmma_kernel(float* __restrict__ AGG,
                                     const float* __restrict__ Wh, int n_node) {
    int wave  = (int)((blockIdx.x * blockDim.x + threadIdx.x) >> 5);
    int lane  = threadIdx.x & 31;
    int m0    = wave << 4;
    if (m0 >= n_node) return;
    int row16 = lane & 15;
    int hi    = lane >> 4;
    int arow_i = m0 + row16;
    if (arow_i >= n_node) arow_i = n_node - 1;
    const float* arow = AGG + (long)arow_i * IN_DIM + 2 * hi;
    const float* brow = Wh  + (long)row16  * IN_DIM + 2 * hi;

    v8f c0 = {}, c1 = {}, c2 = {}, c3 = {};
    #pragma unroll
    for (int kk = 0; kk < IN_DIM / 4; ++kk) {
        v2f a  = *(const v2f*)(arow + 4 * kk);
        v2f b0 = *(const v2f*)(brow + 0 * 16 * IN_DIM + 4 * kk);
        v2f b1 = *(const v2f*)(brow + 1 * 16 * IN_DIM + 4 * kk);
        v2f b2 = *(const v2f*)(brow + 2 * 16 * IN_DIM + 4 * kk);
        v2f b3 = *(const v2f*)(brow + 3 * 16 * IN_DIM + 4 * kk);
        c0 = __builtin_amdgcn_wmma_f32_16x16x4_f32(false, a, false, b0,
                                                   (short)0, c0, false, false);
        c1 = __builtin_amdgcn_wmma_f32_16x16x4_f32(false, a, false, b1,
                                                   (short)0, c1, false, false);
        c2 = __builtin_amdgcn_wmma_f32_16x16x4_f32(false, a, false, b2,
                                                   (short)0, c2, false, false);
        c3 = __builtin_amdgcn_wmma_f32_16x16x4_f32(false, a, false, b3,
                                                   (short)0, c3, false, false);
    }
    #pragma unroll
    for (int j = 0; j < 8; ++j) {
        int m = m0 + j + 8 * hi;
        if (m < n_node) {
            float* o = AGG + (long)m * OUT_DIM;
            o[row16]      = c0[j];
            o[16 + row16] = c1[j];
            o[32 + row16] = c2[j];
            o[48 + row16] = c3[j];
        }
    }
}

// ---------------------------------------------------------------------------
extern "C" void kernel_launch(void* const* d_in, const int* in_sizes, int n_in,
                              void* d_out, int out_size, void* d_ws, size_t ws_size,
                              hipStream_t stream) {
    // setup_inputs order:
    // 0 q_sub, 1 q_rel, 2 hidden, 3 edges, 4 nodes, 5 old_nodes_new_idx,
    // 6 batchsize, 7 rela_embed, 8 Ws, 9 Wr, 10 Wqr, 11 bqr,
    // 12 w_alpha_w, 13 w_alpha_b, 14 Wh
    const int*   q_rel     = (const int*)  d_in[1];
    const float* hidden    = (const float*)d_in[2];
    const int*   edges     = (const int*)  d_in[3];
    const float* rela      = (const float*)d_in[7];
    const float* Ws        = (const float*)d_in[8];
    const float* Wr        = (const float*)d_in[9];
    const float* Wqr       = (const float*)d_in[10];
    const float* bqr       = (const float*)d_in[11];
    const float* w_alpha_w = (const float*)d_in[12];
    const float* w_alpha_b = (const float*)d_in[13];
    const float* Wh        = (const float*)d_in[14];

    const int n_node = in_sizes[2] / IN_DIM;
    const int n_edge = in_sizes[3] / 6;
    const int nrel   = in_sizes[7] / IN_DIM;
    const int batch  = in_sizes[1];

    float* agg     = (float*)d_out;                       // reused as segment-sum buffer
    float* S       = (float*)d_ws;                        // n_node x 32
    float* attn_r  = S + (long)n_node * ATTN_DIM;         // nrel x 32
    float* attn_qr = attn_r + (long)nrel * ATTN_DIM;      // batch x 32

    // 1) zero aggregation buffer (d_out)
    long n4 = (long)n_node * OUT_DIM / 4;
    zero_f4_kernel<<<(int)((n4 + 255) / 256), 256, 0, stream>>>((float4*)agg, n4);

    // 2) per-relation / per-query attention precompute
    int pre_waves = nrel + batch;
    attn_pre_kernel<<<(pre_waves + 7) / 8, 256, 0, stream>>>(
        rela, Wr, Wqr, bqr, q_rel, attn_r, attn_qr, nrel, batch);

    // 3) S = hidden @ Ws.T via f32 WMMA
    int mtiles = (n_node + 15) / 16;
    node_attn_wmma_kernel<<<(mtiles + 7) / 8, 256, 0, stream>>>(hidden, Ws, S, n_node);

    // 4) edge pass: gather + gate + atomic scatter-add (dominant, L2-bound)
    edge_kernel<<<(n_edge + 7) / 8, 256, 0, stream>>>(
        edges, hidden, rela, S, attn_r, attn_qr, w_alpha_w, w_alpha_b, agg, n_edge);

    // 5) hidden_new = agg @ Wh.T in-place on d_out via f32 WMMA
    out_gemm_wmma_kernel<<<(mtiles + 7) / 8, 256, 0, stream>>>(agg, Wh, n_node);
}